// CriticNetwork_49546742726783
// MI455X (gfx1250) — compile-verified
//
#include <hip/hip_runtime.h>
#include <hip/hip_bf16.h>

#define DD 128   // feature dim
#define HH 32    // hidden dim

typedef __attribute__((ext_vector_type(2))) float v2f;
typedef __attribute__((ext_vector_type(8))) float v8f;

// ---------------------------------------------------------------------------
// K0: init workspace: deg = 1.0 (self-loop), agg = 0, gsum = 0
// ---------------------------------------------------------------------------
__global__ void gcn_init_kernel(float* __restrict__ deg, float* __restrict__ agg,
                                float* __restrict__ gsum, int N, long long totalND) {
    long long i = (long long)blockIdx.x * blockDim.x + threadIdx.x;
    if (i < totalND) agg[i] = 0.0f;
    if (i < N)       deg[i] = 1.0f;       // self-loop contributes 1 to degree
    if (i < DD)      gsum[i] = 0.0f;
}

// ---------------------------------------------------------------------------
// K1: deg[dst[e]] += 1 over real edges (edge list streamed once -> NT loads)
// ---------------------------------------------------------------------------
__global__ void gcn_degree_kernel(const int* __restrict__ dst, float* __restrict__ deg, int E) {
    int e = blockIdx.x * blockDim.x + threadIdx.x;
    if (e < E) {
        int d = __builtin_nontemporal_load(&dst[e]);
        atomicAdd(&deg[d], 1.0f);
    }
}

// ---------------------------------------------------------------------------
// K2: deg -> dinv = rsqrt(deg) in place (deg >= 1 always, keep guard anyway)
// ---------------------------------------------------------------------------
__global__ void gcn_dinv_kernel(float* __restrict__ deg, int N) {
    int i = blockIdx.x * blockDim.x + threadIdx.x;
    if (i < N) {
        float d = deg[i];
        deg[i] = (d > 0.0f) ? rsqrtf(d) : 0.0f;
    }
}

// ---------------------------------------------------------------------------
// K3: xw = x @ W_gcn  via V_WMMA_F32_16X16X4_F32 (exact f32 math)
// Block = 256 threads = 8 waves; wave w of block b owns tile (b, w) of the
// [N/16, 8] tile grid (8 = 128/16 column tiles). Full W (64 KB) staged in LDS.
// A-frag layout (32-bit A 16x4): lanes 0-15 -> K+0/K+1, lanes 16-31 -> K+2/K+3.
// C/D layout: VGPR j: lanes 0-15 -> M=j, lanes 16-31 -> M=j+8; N = lane&15.
// ---------------------------------------------------------------------------
__global__ void __launch_bounds__(256)
gcn_gemm_xw_kernel(const float* __restrict__ x, const float* __restrict__ W,
                   float* __restrict__ xw, int nrows) {
    __shared__ float Ws[DD * DD];   // 64 KB (WGP LDS = 320 KB)
    // cooperative load of the whole 128x128 weight matrix, float4-wide
    const float4* W4 = (const float4*)W;
    float4* Ws4 = (float4*)Ws;
    for (int i = threadIdx.x; i < DD * DD / 4; i += blockDim.x) Ws4[i] = W4[i];
    __syncthreads();

    const int wave = threadIdx.x >> 5;
    const int lane = threadIdx.x & 31;
    const int tile = blockIdx.x * 8 + wave;
    const int rowTile = tile >> 3;       // 8 column tiles (128/16)
    const int colTile = tile & 7;
    const int row0 = rowTile * 16;
    const int col0 = colTile * 16;
    if (row0 >= nrows) return;           // wave-uniform

    const int l    = lane & 15;          // M (for A) / N (for B,C,D)
    const int kgrp = (lane >> 4) * 2;    // K sub-group offset: 0 or 2

    int arow = row0 + l;
    if (arow >= nrows) arow = nrows - 1;       // clamp tail loads (keeps EXEC full)
    const float* __restrict__ xrow = x + (long long)arow * DD;

    v8f c = {};
    #pragma unroll 4
    for (int kk = 0; kk < DD; kk += 4) {
        v2f a, b;
        a.x = xrow[kk + kgrp + 0];
        a.y = xrow[kk + kgrp + 1];
        b.x = Ws[(kk + kgrp + 0) * DD + col0 + l];
        b.y = Ws[(kk + kgrp + 1) * DD + col0 + l];
        // 8 args: (neg_a, A, neg_b, B, c_mod, C, reuse_a, reuse_b)
        c = __builtin_amdgcn_wmma_f32_16x16x4_f32(
                false, a, false, b, (short)0, c, false, false);
    }

    const int mbase = row0 + (lane >> 4) * 8;
    const int col   = col0 + l;
    float* __restrict__ outp = xw + (long long)mbase * DD + col;
    if (row0 + 16 <= nrows) {
        // full tile (the only case when N % 16 == 0): clean 8-store clause,
        // each store is two 64B contiguous half-wave segments.
        #pragma unroll
        for (int j = 0; j < 8; ++j) outp[(long long)j * DD] = c[j];
    } else {
        #pragma unroll
        for (int j = 0; j < 8; ++j) {
            if (mbase + j < nrows) outp[(long long)j * DD] = c[j];
        }
    }
}

// ---------------------------------------------------------------------------
// K4: edge scatter: agg[dst] += dinv[src]*dinv[dst] * xw[src]
// One wave per edge: 32 lanes x float4 = 128 channels, coalesced row gather,
// scatter via global_atomic_add_f32 (L2-resident: agg is 25.6 MB << 192 MB L2).
// Edge lists are streamed once -> non-temporal loads so they don't evict the
// L2-resident xw/agg working set.
// ---------------------------------------------------------------------------
__global__ void __launch_bounds__(256)
gcn_edge_agg_kernel(const int* __restrict__ src, const int* __restrict__ dst,
                    const float* __restrict__ dinv, const float* __restrict__ xw,
                    float* __restrict__ agg, int E) {
    const int wid  = (int)(((long long)blockIdx.x * blockDim.x + threadIdx.x) >> 5);
    const int lane = threadIdx.x & 31;
    if (wid >= E) return;

    // lookahead prefetch of a future edge's source row (global_prefetch_b8)
    const int wid2 = wid + 64;
    if (wid2 < E) {
        const float4* xs2 = (const float4*)(xw + (long long)__builtin_nontemporal_load(&src[wid2]) * DD);
        __builtin_prefetch(xs2 + lane, 0, 0);
    }

    const int s = __builtin_nontemporal_load(&src[wid]);
    const int d = __builtin_nontemporal_load(&dst[wid]);
    const float norm = dinv[s] * dinv[d];

    const float4* xs = (const float4*)(xw + (long long)s * DD);
    float4 v = xs[lane];
    float* ad = agg + (long long)d * DD + lane * 4;
    atomicAdd(ad + 0, v.x * norm);
    atomicAdd(ad + 1, v.y * norm);
    atomicAdd(ad + 2, v.z * norm);
    atomicAdd(ad + 3, v.w * norm);
}

// ---------------------------------------------------------------------------
// K5: conv = agg + dinv^2*xw (self-loop) + b ; h = relu(conv) + x ;
//     gsum[ch] += sum_rows h   (blockDim = 128 = one thread per channel)
// ---------------------------------------------------------------------------
__global__ void __launch_bounds__(128)
gcn_node_reduce_kernel(const float* __restrict__ x, const float* __restrict__ xw,
                       const float* __restrict__ agg, const float* __restrict__ dinv,
                       const float* __restrict__ b_gcn, float* __restrict__ gsum, int N) {
    const int ch = threadIdx.x;
    const float bch = b_gcn[ch];
    float acc = 0.0f;
    for (int row = blockIdx.x; row < N; row += gridDim.x) {
        const float di = dinv[row];
        const long long off = (long long)row * DD + ch;
        float conv = agg[off] + di * di * xw[off] + bch;
        float h = fmaxf(conv, 0.0f) + x[off];
        acc += h;
    }
    atomicAdd(&gsum[ch], acc);
}

// ---------------------------------------------------------------------------
// K6: tiny MLP head: 128 -> 32 -> 32 -> 1 (single block, 32 threads)
// ---------------------------------------------------------------------------
__global__ void __launch_bounds__(32)
gcn_mlp_kernel(const float* __restrict__ gsum,
               const float* __restrict__ W1, const float* __restrict__ b1,
               const float* __restrict__ W2, const float* __restrict__ b2,
               const float* __restrict__ W3, const float* __restrict__ b3,
               float* __restrict__ out) {
    __shared__ float g1[HH];
    __shared__ float g2[HH];
    const int t = threadIdx.x;  // 0..31

    float s = b1[t];
    #pragma unroll 4
    for (int d = 0; d < DD; ++d) s += gsum[d] * W1[d * HH + t];
    g1[t] = fmaxf(s, 0.0f);
    __syncthreads();

    float s2 = b2[t];
    #pragma unroll
    for (int i = 0; i < HH; ++i) s2 += g1[i] * W2[i * HH + t];
    g2[t] = fmaxf(s2, 0.0f);
    __syncthreads();

    if (t == 0) {
        float o = b3[0];
        #pragma unroll
        for (int i = 0; i < HH; ++i) o += g2[i] * W3[i];
        out[0] = o;
    }
}

// ---------------------------------------------------------------------------
extern "C" void kernel_launch(void* const* d_in, const int* in_sizes, int n_in,
                              void* d_out, int out_size, void* d_ws, size_t ws_size,
                              hipStream_t stream) {
    const float* x  = (const float*)d_in[0];
    const int*   ei = (const int*)d_in[1];     // [2, E] row-major
    const float* Wg = (const float*)d_in[2];
    const float* bg = (const float*)d_in[3];
    const float* W1 = (const float*)d_in[4];
    const float* b1 = (const float*)d_in[5];
    const float* W2 = (const float*)d_in[6];
    const float* b2 = (const float*)d_in[7];
    const float* W3 = (const float*)d_in[8];
    const float* b3 = (const float*)d_in[9];

    const int N = in_sizes[0] / DD;
    const int E = in_sizes[1] / 2;
    const int* src = ei;          // edge_index[0] = message sources
    const int* dst = ei + E;      // edge_index[1] = aggregation targets

    // workspace layout (floats): deg/dinv[N] | xw[N*D] | agg[N*D] | gsum[D]
    float* ws   = (float*)d_ws;
    float* deg  = ws;
    float* xw   = deg + N;
    float* agg  = xw + (long long)N * DD;
    float* gsum = agg + (long long)N * DD;

    const long long totalND = (long long)N * DD;
    {
        long long blocks = (totalND + 255) / 256;
        gcn_init_kernel<<<(int)blocks, 256, 0, stream>>>(deg, agg, gsum, N, totalND);
    }
    gcn_degree_kernel<<<(E + 255) / 256, 256, 0, stream>>>(dst, deg, E);
    gcn_dinv_kernel<<<(N + 255) / 256, 256, 0, stream>>>(deg, N);

    const int rowTiles = (N + 15) / 16;
    const int tiles = rowTiles * 8;                   // 8 column tiles of 16
    gcn_gemm_xw_kernel<<<(tiles + 7) / 8, 256, 0, stream>>>(x, Wg, xw, N);

    {
        long long threads = (long long)E * 32;        // one wave per edge
        long long blocks = (threads + 255) / 256;
        gcn_edge_agg_kernel<<<(int)blocks, 256, 0, stream>>>(src, dst, deg, xw, agg, E);
    }

    gcn_node_reduce_kernel<<<512, 128, 0, stream>>>(x, xw, agg, deg, bg, gsum, N);
    gcn_mlp_kernel<<<1, 32, 0, stream>>>(gsum, W1, b1, W2, b2, W3, b3, (float*)d_out);
}